// AttentionLayer_bkp_46712064311860
// MI455X (gfx1250) — compile-verified
//
#include <hip/hip_runtime.h>

// Problem dims (fixed by the reference).
#define T_DIM 2048
#define B_DIM 16
#define H_DIM 1024
#define WIN   3
#define ROWS  (T_DIM * B_DIM)   // 32768 rows of X, row-major stride H

typedef __attribute__((ext_vector_type(2))) float v2f;
typedef __attribute__((ext_vector_type(8))) float v8f;

#define KC      32               // K-chunk staged per LDS fill
#define NB      8                // accumulator N-tiles per wave (8*16 = 128 cols)
#define MWAVES  8                // waves per block -> 128 M rows per block
// K-pair-interleaved LDS tile: element (k,col) at (k>>1)*PITCH + 2*col + (k&1).
// PITCH = 2*128 + 32 words; PITCH % 64 == 32 -> the two half-wave pair-rows hit
// disjoint bank halves, so the 8-byte B-fragment loads are conflict-free.
#define PITCH   288
#define NCHUNKS ((H_DIM / (NB * 16)) * (H_DIM / KC))   // 8 N-blocks * 32 K-chunks = 256

// ---------------------------------------------------------------------------
// Kernel 1: scores[r] = tanh(X[r,:] @ W) @ proj   (fused epilogue, no 128MB tmp)
// W tiles staged into LDS with gfx1250 async-LDS copies (double-buffered, copy
// hides under the 64 WMMAs of the current chunk); B fragments are single
// aligned ds_load_b64 straight into the WMMA register pair (no repacking).
// ---------------------------------------------------------------------------
__global__ __launch_bounds__(256) void scores_kernel(const float* __restrict__ X,
                                                     const float* __restrict__ Wm,
                                                     const float* __restrict__ proj,
                                                     float* __restrict__ scores) {
    __shared__ float ldsW[2][(KC / 2) * PITCH];

    const int tid      = threadIdx.x;
    const int wave     = tid >> 5;
    const int lane     = tid & 31;
    const int lane16   = lane & 15;     // A: M index / B,C,D: N column within tile
    const int laneHalf = lane >> 4;     // selects K pair (A/B) and M half (C/D)
    const int mbase    = blockIdx.x * (MWAVES * 16);
    const int rowM     = mbase + wave * 16 + lane16;
    const float* xrow  = X + (size_t)rowM * H_DIM;

    // Async stage of one 32x128 W tile (16KB) into the interleaved layout:
    // 4096 x b32, 16 per thread. VDST VGPR = LDS byte address (low 32 bits of
    // the flat shared pointer), VADDR = 64-bit global address. ASYNCcnt-tracked.
    auto prefetch = [&](int ci, int buf) {
        const int nbase_c = (ci >> 5) * (NB * 16);
        const int kc_c    = (ci & 31) * KC;
#pragma unroll
        for (int j = 0; j < 16; ++j) {
            const int e = tid + 256 * j;          // 0..4095
            const int r = e >> 7;                 // 0..31  (K row)
            const int c = e & 127;                // 0..127 (N col)
            const float*   g   = Wm + (size_t)(kc_c + r) * H_DIM + nbase_c + c;
            const unsigned lds =
                (unsigned)(size_t)&ldsW[buf][(r >> 1) * PITCH + 2 * c + (r & 1)];
            asm volatile("global_load_async_to_lds_b32 %0, %1, off"
                         :: "v"(lds), "v"(g) : "memory");
        }
    };

    const v8f vzero = {};
    float s[8];
#pragma unroll
    for (int v = 0; v < 8; ++v) s[v] = 0.0f;

    int cur = 0;
    prefetch(0, 0);
    asm volatile("s_wait_asynccnt 0x0" ::: "memory");
    __syncthreads();

    for (int nb = 0; nb < H_DIM / (NB * 16); ++nb) {
        const int nbase = nb * (NB * 16);

        v8f acc[NB];
#pragma unroll
        for (int nt = 0; nt < NB; ++nt) acc[nt] = vzero;

        for (int kcI = 0; kcI < H_DIM / KC; ++kcI) {
            const int ci = (nb << 5) | kcI;
            const int kc = kcI * KC;

            // Kick off the next tile's async copies into the other buffer.
            if (ci + 1 < NCHUNKS) prefetch(ci + 1, cur ^ 1);

            // A fragments for this K-chunk: 16x4 fp32 layout wants
            // lane<16 : {K=4k, 4k+1}, lane>=16 : {K=4k+2, 4k+3} for row M=lane16.
            v2f a[KC / 4];
#pragma unroll
            for (int ks = 0; ks < KC / 4; ++ks)
                a[ks] = *(const v2f*)(xrow + kc + 4 * ks + 2 * laneHalf);

            const float* wbuf = ldsW[cur];
#pragma unroll
            for (int nt = 0; nt < NB; ++nt) {
                const int col2 = 2 * (nt * 16 + lane16);
#pragma unroll
                for (int ks = 0; ks < KC / 4; ++ks) {
                    // rows {4ks+2h, 4ks+2h+1} for this lane's col: one b64 load.
                    const v2f b =
                        *(const v2f*)&wbuf[(2 * ks + laneHalf) * PITCH + col2];
                    // D = A(16x4,f32) * B(4x16,f32) + C(16x16,f32)
                    acc[nt] = __builtin_amdgcn_wmma_f32_16x16x4_f32(
                        false, a[ks], false, b, (short)0, acc[nt], false, false);
                }
            }

            // Own prefetch complete, then everyone done consuming -> swap.
            asm volatile("s_wait_asynccnt 0x0" ::: "memory");
            __syncthreads();
            cur ^= 1;
        }

        // Fused epilogue: s[m] += tanh(Y[m,n]) * proj[n] for this 128-col block.
#pragma unroll
        for (int nt = 0; nt < NB; ++nt) {
            const float p = proj[nbase + nt * 16 + lane16];
#pragma unroll
            for (int v = 0; v < 8; ++v)
                s[v] += tanhf(acc[nt][v]) * p;
        }
    }

    // Reduce the partial scores across the 16 lanes of each half
    // (xor masks 1,2,4,8 never cross the lane16 boundary).
#pragma unroll
    for (int v = 0; v < 8; ++v) {
        float x = s[v];
        x += __shfl_xor(x, 1, 32);
        x += __shfl_xor(x, 2, 32);
        x += __shfl_xor(x, 4, 32);
        x += __shfl_xor(x, 8, 32);
        s[v] = x;
    }
    if (lane16 == 0) {
        const int base = mbase + wave * 16 + laneHalf * 8; // M=v (half 0) / M=v+8 (half 1)
#pragma unroll
        for (int v = 0; v < 8; ++v) scores[base + v] = s[v];
    }
}

// ---------------------------------------------------------------------------
// Kernel 2: sliding-window (w=3) softmax attend. One block per (t,b) row.
// Pure bandwidth: X mostly resident in 192MB L2 after kernel 1.
// ---------------------------------------------------------------------------
__global__ __launch_bounds__(256) void attend_kernel(const float* __restrict__ X,
                                                     const float* __restrict__ scores,
                                                     float* __restrict__ out) {
    const int rb = blockIdx.x;          // r = t*B + b
    const int t  = rb / B_DIM;
    const size_t rowOff = (size_t)rb * H_DIM;
    const int h4 = threadIdx.x << 2;    // float4 lane offset within H

    if (t < WIN) {                      // pass-through for the first w steps
        const float4 v = *(const float4*)(X + rowOff + h4);
        *(float4*)(out + rowOff + h4) = v;
        return;
    }

    const int b = rb - t * B_DIM;
    const float s0 = scores[(t - 3) * B_DIM + b];
    const float s1 = scores[(t - 2) * B_DIM + b];
    const float s2 = scores[(t - 1) * B_DIM + b];
    const float m  = fmaxf(s0, fmaxf(s1, s2));
    float e0 = __expf(s0 - m), e1 = __expf(s1 - m), e2 = __expf(s2 - m);
    const float inv = 1.0f / (e0 + e1 + e2);
    e0 *= inv; e1 *= inv; e2 *= inv;

    const size_t stride = (size_t)B_DIM * H_DIM;
    const float4 x0 = *(const float4*)(X + rowOff - 3 * stride + h4);
    const float4 x1 = *(const float4*)(X + rowOff - 2 * stride + h4);
    const float4 x2 = *(const float4*)(X + rowOff - 1 * stride + h4);

    float4 o;
    o.x = fmaf(e0, x0.x, fmaf(e1, x1.x, e2 * x2.x));
    o.y = fmaf(e0, x0.y, fmaf(e1, x1.y, e2 * x2.y));
    o.z = fmaf(e0, x0.z, fmaf(e1, x1.z, e2 * x2.z));
    o.w = fmaf(e0, x0.w, fmaf(e1, x1.w, e2 * x2.w));
    *(float4*)(out + rowOff + h4) = o;
}

// ---------------------------------------------------------------------------
extern "C" void kernel_launch(void* const* d_in, const int* in_sizes, int n_in,
                              void* d_out, int out_size, void* d_ws, size_t ws_size,
                              hipStream_t stream) {
    (void)in_sizes; (void)n_in; (void)out_size; (void)ws_size;
    const float* X    = (const float*)d_in[0];   // [T,B,H] fp32
    const float* Wm   = (const float*)d_in[1];   // [H,H]   fp32
    const float* proj = (const float*)d_in[2];   // [H,1]   fp32
    // d_in[3] = attention_width (= 3, static in the reference)
    float* scores = (float*)d_ws;                // ROWS floats = 128 KB scratch
    float* out    = (float*)d_out;               // [T,B,H] fp32

    scores_kernel<<<ROWS / (MWAVES * 16), 256, 0, stream>>>(X, Wm, proj, scores);
    attend_kernel<<<ROWS, 256, 0, stream>>>(X, scores, out);
}